// BaseGenerator_4853313044689
// MI455X (gfx1250) — compile-verified
//
#include <hip/hip_runtime.h>

typedef __attribute__((ext_vector_type(2))) float v2f;
typedef __attribute__((ext_vector_type(8))) float v8f;

#define ROWS_P1 16384   // output pixels (rows of mapping)
#define K_DIM   16384   // base pixels (cols of mapping)
#define WAVES_PER_BLOCK 8
#define K_PER_WAVE (K_DIM / WAVES_PER_BLOCK)   // 2048

// ---------------------------------------------------------------------------
// Copy base_image -> first 49152 floats of d_out
// ---------------------------------------------------------------------------
__global__ void copy_base_kernel(const float* __restrict__ base,
                                 float* __restrict__ out, int n) {
    int i = blockIdx.x * blockDim.x + threadIdx.x;
    if (i < n) out[i] = base[i];
}

// ---------------------------------------------------------------------------
// mapped = mapping @ base_flat using V_WMMA_F32_16X16X4_F32.
// One block = 16 output rows. 8 waves split the K=16384 reduction, each wave
// runs 512 WMMAs (K step 4), partials are reduced through LDS.
// ---------------------------------------------------------------------------
__global__ __launch_bounds__(256) void remap_wmma_kernel(
    const float* __restrict__ mapping,   // [16384, 16384] row-major
    const float* __restrict__ base,      // [16384, 3]
    float* __restrict__ out)             // [98304]; mapped goes at +49152
{
    __shared__ float red[WAVES_PER_BLOCK * 256];

    const int tid   = threadIdx.x;
    const int lane  = tid & 31;
    const int wave  = tid >> 5;
    const int r0    = blockIdx.x * 16;        // first output row of this tile
    const int kbeg  = wave * K_PER_WAVE;      // this wave's K slice

    // ISA 32-bit A 16x4 layout: lanes 0-15 hold K={0,1}, lanes 16-31 K={2,3}
    const int arow  = lane & 15;              // M within tile
    const int khalf = lane >> 4;              // 0 or 1
    const float* aptr = mapping + (size_t)(r0 + arow) * K_DIM + kbeg + 2 * khalf;

    // B (4x16, K x N): VGPR0 = rows K0/K2, VGPR1 = rows K1/K3 across lanes.
    // Columns 3..15 are dummies: clamp to column 0 (values never stored out),
    // keeping the load unconditional so EXEC is all-ones at every WMMA.
    const int ncol  = lane & 15;
    const int bc    = (ncol < 3) ? ncol : 0;
    const float* bptr = base + (size_t)(kbeg + 2 * khalf) * 3 + bc;

    v8f c = {0.f, 0.f, 0.f, 0.f, 0.f, 0.f, 0.f, 0.f};

    #pragma unroll 4
    for (int kk = 0; kk < K_PER_WAVE; kk += 4) {
        // Streaming read of mapping: non-temporal so the 1 GiB one-shot
        // stream does not evict the reused 196 KB base matrix from L2.
        v2f a = __builtin_nontemporal_load((const v2f*)aptr);
        v2f b;
        b.x = bptr[0];   // row K = kbeg+kk+2*khalf
        b.y = bptr[3];   // row K = kbeg+kk+2*khalf+1
        c = __builtin_amdgcn_wmma_f32_16x16x4_f32(
                /*neg_a=*/false, a, /*neg_b=*/false, b,
                /*c_mod=*/(short)0, c, /*reuse_a=*/false, /*reuse_b=*/false);
        aptr += 4;
        bptr += 12;
    }

    // Spill this wave's 16x16 partial to LDS (lane-major, 8 floats per lane).
    #pragma unroll
    for (int v = 0; v < 8; ++v)
        red[wave * 256 + lane * 8 + v] = c[v];

    __syncthreads();

    // 8-way split-K reduction: thread t owns (lane = t>>3, vgpr = t&7).
    float s = red[tid];
    #pragma unroll
    for (int w = 1; w < WAVES_PER_BLOCK; ++w)
        s += red[w * 256 + tid];

    const int sl = tid >> 3;                 // original lane 0..31
    const int sv = tid & 7;                  // original VGPR 0..7
    const int m  = sv + ((sl >> 4) << 3);    // C/D layout: M = v + 8*(lane>=16)
    const int n  = sl & 15;                  // N = lane & 15
    if (n < 3)
        out[(size_t)ROWS_P1 * 3 + (size_t)(r0 + m) * 3 + n] = s;
}

// ---------------------------------------------------------------------------
extern "C" void kernel_launch(void* const* d_in, const int* in_sizes, int n_in,
                              void* d_out, int out_size, void* d_ws, size_t ws_size,
                              hipStream_t stream) {
    const float* mapping = (const float*)d_in[0];   // 16384*16384 f32
    const float* base    = (const float*)d_in[1];   // 128*128*3 = 49152 f32
    float* out = (float*)d_out;                     // 98304 f32

    const int base_n = 49152;
    copy_base_kernel<<<(base_n + 255) / 256, 256, 0, stream>>>(base, out, base_n);

    remap_wmma_kernel<<<ROWS_P1 / 16, 256, 0, stream>>>(mapping, base, out);
}